// BeliefDynamics_55731495633058
// MI455X (gfx1250) — compile-verified
//
#include <hip/hip_runtime.h>
#include <math.h>

// ---------------------------------------------------------------------------
// BeliefDynamics on MI455X (gfx1250)
//   new_mean : closed-form linear gradient + norm reduction      (O(D))
//   new_cov  : scalar * covariance (Hessian is constant -101*I)  (O(D^2))
//   precision: Newton-Schulz SPD inverse, pure f32 WMMA GEMMs    (O(iters*D^3))
// ---------------------------------------------------------------------------

#define NMAT 1024
#define NN   (NMAT * NMAT)
#define KC   64          // k-chunk staged in LDS per block
#define CN   32          // C columns per wave (two 16x16 WMMA subtiles)

typedef __attribute__((ext_vector_type(2))) float v2f;
typedef __attribute__((ext_vector_type(8))) float v8f;

// ---------------------------------------------------------------------------
// Kernel 1: new_mean (single block of 1024 threads; LDS reduction for ||g||)
// g = -100*(mean-obs) - mean + score
// new_mean = mean - 0.1*g*dt_adapt + sqrt(2*0.1*0.01)*0.1*noise
// ---------------------------------------------------------------------------
__global__ __launch_bounds__(1024) void bd_mean_kernel(
    const float* __restrict__ mean, const float* __restrict__ obs,
    const float* __restrict__ score, const float* __restrict__ noise,
    float* __restrict__ out_mean)
{
    __shared__ float red[1024];
    const int i = threadIdx.x;
    const float m = mean[i];
    const float g = -100.0f * (m - obs[i]) - m + score[i];
    red[i] = g * g;
    __syncthreads();
    #pragma unroll
    for (int s = 512; s > 0; s >>= 1) {
        if (i < s) red[i] += red[i + s];
        __syncthreads();
    }
    const float gnorm = sqrtf(red[0]);
    const float adt = 0.01f / (1.0f + 0.1f * gnorm);
    // sqrt(2*0.1*0.01)*0.1 = sqrt(0.002)*0.1
    out_mean[i] = m - 0.1f * g * adt + 0.0044721359549995794f * noise[i];
}

// ---------------------------------------------------------------------------
// Kernel 2: out_cov = c*cov ; M = c*cov + 1e-8*I  (c = exp(4.044))
// ---------------------------------------------------------------------------
__global__ __launch_bounds__(256) void bd_scale_kernel(
    const float* __restrict__ cov, float* __restrict__ out_cov,
    float* __restrict__ M, float c)
{
    const int idx = blockIdx.x * 256 + threadIdx.x;
    const float v = c * cov[idx];
    out_cov[idx] = v;
    const int r = idx >> 10, cc = idx & (NMAT - 1);
    M[idx] = v + ((r == cc) ? 1e-8f : 0.0f);
}

// ---------------------------------------------------------------------------
// Kernel 3a: zero the row-sum scalar (ws is poisoned; must init every call)
// ---------------------------------------------------------------------------
__global__ void bd_init_kernel(unsigned* rbits) { *rbits = 0u; }

// ---------------------------------------------------------------------------
// Kernel 3b: r = max_row sum |M[row][:]|  (upper bound on lambda_max, SPD)
// positive floats -> uint bit pattern is order-preserving -> atomicMax works
// ---------------------------------------------------------------------------
__global__ __launch_bounds__(256) void bd_rowmax_kernel(
    const float* __restrict__ M, unsigned* __restrict__ rbits)
{
    __shared__ float red[256];
    const int row = blockIdx.x, t = threadIdx.x;
    float s = 0.0f;
    #pragma unroll
    for (int j = t; j < NMAT; j += 256) s += fabsf(M[row * NMAT + j]);
    red[t] = s;
    __syncthreads();
    #pragma unroll
    for (int k = 128; k > 0; k >>= 1) {
        if (t < k) red[t] += red[t + k];
        __syncthreads();
    }
    if (t == 0) atomicMax(rbits, __float_as_uint(red[0]));
}

// ---------------------------------------------------------------------------
// Kernel 3c: X0 = (1/r) * I
// ---------------------------------------------------------------------------
__global__ __launch_bounds__(256) void bd_setx0_kernel(
    float* __restrict__ X, const unsigned* __restrict__ rbits)
{
    const float inv = 1.0f / __uint_as_float(*rbits);
    const int idx = blockIdx.x * 256 + threadIdx.x;
    const int r = idx >> 10, cc = idx & (NMAT - 1);
    X[idx] = (r == cc) ? inv : 0.0f;
}

// ---------------------------------------------------------------------------
// Kernel 4: C = alpha*(A@B) + beta*D  via V_WMMA_F32_16X16X4_F32
//   block = 256 threads = 8 waves; wave w -> 16x32 C tile:
//     rows  (blockIdx.x*8 + w)*16 .. +16
//     cols  blockIdx.y*32 .. +32      (two 16-wide WMMA subtiles)
//   - One A fragment reused across two B fragments -> two independent
//     accumulator chains, half the A traffic per FLOP.
//   - LDS holds B pair-interleaved: sB[(p*CN + c)*2 + e] = B[kb+2p+e][cb+c],
//     so each B fragment is ONE aligned ds_load_b64 directly into the WMMA
//     operand VGPR pair (no fix-up movs), and all 16 k-step loads share one
//     base address with immediate offsets (clauseable).
//   - All chunk A fragments preloaded into registers -> one global clause,
//     incremental s_wait counts.
//   A fragment = aligned float2 per ISA 32-bit 16x4 layout
//   (VGPR0 = K0|K2, VGPR1 = K1|K3 across lane halves). Full EXEC everywhere.
// ---------------------------------------------------------------------------
__global__ __launch_bounds__(256) void bd_gemm_kernel(
    float* __restrict__ C, const float* __restrict__ A,
    const float* __restrict__ B, const float* __restrict__ D,
    float alpha, float beta)
{
    __shared__ float sB[KC * CN];   // pair-interleaved layout, 8 KB
    const int tid  = threadIdx.x;
    const int lane = tid & 31;
    const int wave = tid >> 5;
    const int half = lane >> 4;     // 0: K pair {0,1}, 1: K pair {2,3}
    const int mn   = lane & 15;     // A row within tile / B,C col within subtile
    const int tm   = blockIdx.x * 8 + wave;
    const int cb   = blockIdx.y * CN;           // first C column of this block
    const int arow = tm * 16 + mn;              // global A row for this lane
    const float* Arow = A + (size_t)arow * NMAT + 2 * half;

    v8f acc0 = {}, acc1 = {};
    for (int kb = 0; kb < NMAT; kb += KC) {
        // stage B[kb..kb+KC)[cb..cb+CN) pair-interleaved: thread q -> (pair p, col c)
        // contiguous ds_store_b64 across lanes (conflict-free), coalesced b32 reads
        #pragma unroll
        for (int e = 0; e < (KC / 2) * CN / 256; ++e) {     // 4 pairs/thread
            const int q = e * 256 + tid;
            const int p = q >> 5;               // row pair 0..31
            const int c = q & (CN - 1);         // col 0..31
            v2f pr;
            pr.x = B[(kb + 2 * p + 0) * NMAT + cb + c];
            pr.y = B[(kb + 2 * p + 1) * NMAT + cb + c];
            *(v2f*)&sB[(p * CN + c) * 2] = pr;
        }
        __syncthreads();
        if (kb + KC < NMAT)
            __builtin_prefetch(&Arow[kb + KC], 0, 0);
        // preload all A fragments for this chunk (one clause of global_load_b64)
        v2f afrag[KC / 4];
        #pragma unroll
        for (int kk = 0; kk < KC; kk += 4)
            afrag[kk >> 2] = *(const v2f*)&Arow[kb + kk];
        #pragma unroll
        for (int kk = 0; kk < KC; kk += 4) {
            // pair index p = kk/2 + half; fragment = one b64 at immediate offset
            const int pbase = ((kk >> 1) + half) * (CN * 2) + mn * 2;
            v2f b0 = *(const v2f*)&sB[pbase];
            v2f b1 = *(const v2f*)&sB[pbase + 32];
            acc0 = __builtin_amdgcn_wmma_f32_16x16x4_f32(
                false, afrag[kk >> 2], false, b0, (short)0, acc0, false, false);
            acc1 = __builtin_amdgcn_wmma_f32_16x16x4_f32(
                false, afrag[kk >> 2], false, b1, (short)0, acc1, false, false);
        }
        __syncthreads();
    }
    // C/D 16x16 f32 layout: VGPR v, lane half h -> row v + 8*h, col = mn
    #pragma unroll
    for (int v = 0; v < 8; ++v) {
        const int rr = tm * 16 + v + 8 * half;
        const int c0 = cb + mn;
        const int c1 = cb + 16 + mn;
        float o0 = alpha * acc0[v];
        float o1 = alpha * acc1[v];
        if (beta != 0.0f) {
            o0 += beta * D[rr * NMAT + c0];
            o1 += beta * D[rr * NMAT + c1];
        }
        C[rr * NMAT + c0] = o0;
        C[rr * NMAT + c1] = o1;
    }
}

// ---------------------------------------------------------------------------
extern "C" void kernel_launch(void* const* d_in, const int* in_sizes, int n_in,
                              void* d_out, int out_size, void* d_ws, size_t ws_size,
                              hipStream_t stream) {
    (void)in_sizes; (void)n_in; (void)out_size; (void)ws_size;
    const float* mean  = (const float*)d_in[0];
    const float* cov   = (const float*)d_in[1];
    const float* obs   = (const float*)d_in[2];
    const float* score = (const float*)d_in[3];
    const float* noise = (const float*)d_in[4];

    float* out_mean = (float*)d_out;
    float* out_cov  = out_mean + NMAT;
    float* out_prec = out_cov + NN;

    float*    wsf   = (float*)d_ws;
    unsigned* rbits = (unsigned*)wsf;   // scalar slot (64-float pad)
    float*    M     = wsf + 64;         // c*cov + 1e-8*I
    float*    T     = M + NN;           // T = M @ X
    float*    Xa    = T + NN;           // ping buffer; pong = out_prec

    // Covariance drift collapses to a scalar: exp(2 * 202.2 * 0.01)
    const float c = expf(4.044f);

    // 1) new_mean
    bd_mean_kernel<<<1, 1024, 0, stream>>>(mean, obs, score, noise, out_mean);

    // 2) new_cov + M
    bd_scale_kernel<<<NN / 256, 256, 0, stream>>>(cov, out_cov, M, c);

    // 3) X0 = I / (max abs row sum of M)
    bd_init_kernel<<<1, 1, 0, stream>>>(rbits);
    bd_rowmax_kernel<<<NMAT, 256, 0, stream>>>(M, rbits);
    bd_setx0_kernel<<<NN / 256, 256, 0, stream>>>(Xa, rbits);

    // 4) Newton-Schulz: X <- 2X - X(MX); 13 iters (odd) -> result in out_prec
    dim3 ggrid(NMAT / 128, NMAT / CN);  // 8 x 32 blocks, 256 threads each
    for (int it = 0; it < 13; ++it) {
        float* cur = (it & 1) ? out_prec : Xa;
        float* nxt = (it & 1) ? Xa : out_prec;
        bd_gemm_kernel<<<ggrid, 256, 0, stream>>>(T, M, cur, (const float*)nullptr, 1.0f, 0.0f);
        bd_gemm_kernel<<<ggrid, 256, 0, stream>>>(nxt, cur, T, cur, -1.0f, 2.0f);
    }
}